// Model_16475494547949
// MI455X (gfx1250) — compile-verified
//
#include <hip/hip_runtime.h>

typedef __attribute__((ext_vector_type(2))) float v2f;
typedef __attribute__((ext_vector_type(4))) float f4v;
typedef __attribute__((ext_vector_type(8))) float v8f;

#define NN     128          // N = 2 * n_channel
#define HW     16384        // 128*128
#define C1SZ   (16 * HW)    // 262144 floats per activation tensor
#define K2     144          // conv2 im2col depth: 16 ch * 3 * 3
#define SPLIT  32           // split-K factor for fc_c
#define CHUNK  (C1SZ / SPLIT)   // 8192 elements per chunk

// ---------------------------------------------------------------------------
// Kernel 1: degenerate GCN branch.
// Fully-connected graph + self loops => gcn output rows are all identical
// (column mean broadcast). Collapses to a scalar s, then
// g[i] = relu(s * rowsum(fc_g_w[i]) + fc_g_b[i]).
// ---------------------------------------------------------------------------
__global__ __launch_bounds__(128) void graph_branch_k(
    const float* __restrict__ x1, const float* __restrict__ g1w,
    const float* __restrict__ g1b, const float* __restrict__ g2w,
    const float* __restrict__ g2b, const float* __restrict__ fgw,
    const float* __restrict__ fgb, float* __restrict__ g) {
  __shared__ float s_sh;
  if (threadIdx.x == 0) {
    float xm[5];
    for (int f = 0; f < 5; ++f) {
      float a = 0.0f;
      for (int i = 0; i < NN; ++i) a += x1[i * 5 + f];
      xm[f] = a * (1.0f / NN);
    }
    float h1[10];
    for (int k = 0; k < 10; ++k) {
      float a = g1b[k];
      for (int f = 0; f < 5; ++f) a += xm[f] * g1w[k * 5 + f];
      h1[k] = a > 0.0f ? a : 0.0f;
    }
    float s = g2b[0];
    for (int k = 0; k < 10; ++k) s += h1[k] * g2w[k];
    s_sh = s > 0.0f ? s : 0.0f;
  }
  __syncthreads();
  const int i = threadIdx.x;
  float rs = 0.0f;
  for (int j = 0; j < NN; ++j) rs += fgw[i * NN + j];
  const float v = s_sh * rs + fgb[i];
  g[i] = v > 0.0f ? v : 0.0f;
}

// ---------------------------------------------------------------------------
// Kernel 2: conv1 (1 -> 16 channels, 3x3, SAME) + ReLU. 9 MACs/output.
// ---------------------------------------------------------------------------
__global__ __launch_bounds__(256) void conv1_k(
    const float* __restrict__ x2, const float* __restrict__ w1,
    const float* __restrict__ b1, float* __restrict__ c1) {
  const int t = blockIdx.x * 256 + threadIdx.x;     // 0 .. C1SZ-1
  const int o = t >> 14;                            // output channel
  const int p = t & (HW - 1);
  const int y = p >> 7, x = p & 127;
  float a = b1[o];
#pragma unroll
  for (int dy = 0; dy < 3; ++dy) {
#pragma unroll
    for (int dx = 0; dx < 3; ++dx) {
      const int yy = y + dy - 1, xx = x + dx - 1;
      if (yy >= 0 && yy < 128 && xx >= 0 && xx < 128)
        a += w1[o * 9 + dy * 3 + dx] * x2[yy * 128 + xx];
    }
  }
  c1[t] = a > 0.0f ? a : 0.0f;
}

// ---------------------------------------------------------------------------
// Kernel 3: conv2 as im2col GEMM via V_WMMA_F32_16X16X4_F32.
//   D[16 ch x 16 px] += A[16 x 4] * B[4 x 16], 36 K-steps (K = 144).
// A layout: lanes 0-15 -> M=lane, VGPR0/1 = K0/K1; lanes 16-31 -> K2/K3.
// B layout: lanes 0-15 -> N=lane, VGPR0/1 = K0/K1; lanes 16-31 -> K2/K3.
// C/D layout: VGPR v holds M = (lane/16)*8 + v, N = lane%16.
// Branchless padding keeps EXEC all-ones (WMMA requirement).
// ---------------------------------------------------------------------------
__global__ __launch_bounds__(128) void conv2_wmma_k(
    const float* __restrict__ c1, const float* __restrict__ w2,
    const float* __restrict__ b2, float* __restrict__ c2) {
  const int lane  = threadIdx.x & 31;
  const int wave  = threadIdx.x >> 5;
  const int nbase = (blockIdx.x * 4 + wave) * 16;   // 16-pixel tile base
  const int n     = nbase + (lane & 15);
  const int y = n >> 7, x = n & 127;
  const int m     = lane & 15;
  const int khalf = (lane >> 4) * 2;                // 0 or 2

  v8f acc = {};
  for (int k0 = 0; k0 < K2; k0 += 4) {
    const int ka = k0 + khalf;
    v2f a, b;
    a.x = w2[m * K2 + ka];
    a.y = w2[m * K2 + ka + 1];
#pragma unroll
    for (int t = 0; t < 2; ++t) {
      const int kk = ka + t;
      const int ci = kk / 9;
      const int r  = kk - ci * 9;
      const int yy = y + (r / 3) - 1;
      const int xx = x + (r % 3) - 1;
      const bool inb = (yy >= 0) & (yy < 128) & (xx >= 0) & (xx < 128);
      const int idx = ci * HW + (inb ? (yy * 128 + xx) : 0);
      float v = c1[idx];
      v = inb ? v : 0.0f;
      if (t == 0) b.x = v; else b.y = v;
    }
    // (neg_a, A, neg_b, B, c_mod, C, reuse_a, reuse_b)
    acc = __builtin_amdgcn_wmma_f32_16x16x4_f32(
        false, a, false, b, (short)0, acc, false, false);
  }
#pragma unroll
  for (int v = 0; v < 8; ++v) {
    const int mr = (lane >> 4) * 8 + v;
    const float r = acc[v] + b2[mr];
    c2[mr * HW + n] = r > 0.0f ? r : 0.0f;
  }
}

// ---------------------------------------------------------------------------
// Kernel 4: fc_c split-K mat-vec partials. THE bandwidth kernel: streams the
// 128 MB weight matrix once with non-temporal b128 loads (keeps the 1 MB
// activation vector resident in L2). Deterministic two-stage reduction.
// grid = (SPLIT chunks, 128 rows), 256 threads.
// ---------------------------------------------------------------------------
__global__ __launch_bounds__(256) void fc_c_partial_k(
    const float* __restrict__ c2, const float* __restrict__ w,
    float* __restrict__ partials) {
  const int row = blockIdx.y;
  const int chunk = blockIdx.x;
  const int t = threadIdx.x;
  const f4v* __restrict__ wv =
      (const f4v*)(w + (size_t)row * C1SZ + (size_t)chunk * CHUNK);
  const f4v* __restrict__ cv = (const f4v*)(c2 + (size_t)chunk * CHUNK);
  float s = 0.0f;
#pragma unroll
  for (int i = 0; i < CHUNK / 4 / 256; ++i) {       // 8 iterations
    const f4v a = __builtin_nontemporal_load(&wv[t + i * 256]);
    const f4v b = cv[t + i * 256];
    s += a.x * b.x + a.y * b.y + a.z * b.z + a.w * b.w;
  }
  __shared__ float red[256];
  red[t] = s;
  __syncthreads();
  for (int off = 128; off > 0; off >>= 1) {
    if (t < off) red[t] += red[t + off];
    __syncthreads();
  }
  if (t == 0) partials[row * SPLIT + chunk] = red[0];
}

// ---------------------------------------------------------------------------
// Kernel 5: finalize fc_c (+bias+relu) and the fusion head. 1 block, 128 thr.
// ---------------------------------------------------------------------------
__global__ __launch_bounds__(128) void head_k(
    const float* __restrict__ partials, const float* __restrict__ fccb,
    const float* __restrict__ g, const float* __restrict__ fc1w,
    const float* __restrict__ fc1b, const float* __restrict__ fc2w,
    const float* __restrict__ fc2b, const float* __restrict__ fc3w,
    const float* __restrict__ fc3b, float* __restrict__ out) {
  __shared__ float xv[2 * NN];
  __shared__ float y1[NN];
  __shared__ float y2[64];
  const int t = threadIdx.x;
  float a = fccb[t];
  for (int k = 0; k < SPLIT; ++k) a += partials[t * SPLIT + k];
  xv[t] = g[t];
  xv[NN + t] = a > 0.0f ? a : 0.0f;
  __syncthreads();
  float a1 = fc1b[t];
  for (int j = 0; j < 2 * NN; ++j) a1 += fc1w[t * 2 * NN + j] * xv[j];
  y1[t] = a1 > 0.0f ? a1 : 0.0f;
  __syncthreads();
  if (t < 64) {
    float a2 = fc2b[t];
    for (int j = 0; j < NN; ++j) a2 += fc2w[t * NN + j] * y1[j];
    y2[t] = a2 > 0.0f ? a2 : 0.0f;
  }
  __syncthreads();
  if (t == 0) {
    float a3 = fc3b[0];
    for (int j = 0; j < 64; ++j) a3 += fc3w[j] * y2[j];
    out[0] = a3;
  }
}

// ---------------------------------------------------------------------------
extern "C" void kernel_launch(void* const* d_in, const int* in_sizes, int n_in,
                              void* d_out, int out_size, void* d_ws,
                              size_t ws_size, hipStream_t stream) {
  const float* x1      = (const float*)d_in[0];
  const float* x2      = (const float*)d_in[1];
  const float* gcn1_w  = (const float*)d_in[2];
  const float* gcn1_b  = (const float*)d_in[3];
  const float* gcn2_w  = (const float*)d_in[4];
  const float* gcn2_b  = (const float*)d_in[5];
  const float* fc_g_w  = (const float*)d_in[6];
  const float* fc_g_b  = (const float*)d_in[7];
  const float* conv1_w = (const float*)d_in[8];
  const float* conv1_b = (const float*)d_in[9];
  const float* conv2_w = (const float*)d_in[10];
  const float* conv2_b = (const float*)d_in[11];
  const float* fc_c_w  = (const float*)d_in[12];
  const float* fc_c_b  = (const float*)d_in[13];
  const float* fc1_w   = (const float*)d_in[14];
  const float* fc1_b   = (const float*)d_in[15];
  const float* fc2_w   = (const float*)d_in[16];
  const float* fc2_b   = (const float*)d_in[17];
  const float* fc3_w   = (const float*)d_in[18];
  const float* fc3_b   = (const float*)d_in[19];

  float* ws       = (float*)d_ws;
  float* g        = ws;                  // 128
  float* c1       = ws + 128;            // 262144
  float* c2       = c1 + C1SZ;           // 262144
  float* partials = c2 + C1SZ;           // 128 * SPLIT = 4096

  graph_branch_k<<<1, 128, 0, stream>>>(x1, gcn1_w, gcn1_b, gcn2_w, gcn2_b,
                                        fc_g_w, fc_g_b, g);
  conv1_k<<<C1SZ / 256, 256, 0, stream>>>(x2, conv1_w, conv1_b, c1);
  conv2_wmma_k<<<HW / 64, 128, 0, stream>>>(c1, conv2_w, conv2_b, c2);
  dim3 grid4(SPLIT, NN);
  fc_c_partial_k<<<grid4, 256, 0, stream>>>(c2, fc_c_w, partials);
  head_k<<<1, 128, 0, stream>>>(partials, fc_c_b, g, fc1_w, fc1_b, fc2_w,
                                fc2_b, fc3_w, fc3_b, (float*)d_out);
}